// Encoder_36541581754443
// MI455X (gfx1250) — compile-verified
//
#include <hip/hip_runtime.h>
#include <hip/hip_bf16.h>
#include <math.h>

// ---------------------------------------------------------------------------
// MI455X / gfx1250 implementation, fp32 end-to-end (matches fp32 reference;
// the workload is bandwidth-bound so downcasting buys nothing).
// GEMMs: V_WMMA_F32_16X16X4_F32, M-strip per wave (4x B-fragment reuse),
// LDS-tiled activations with bank-conflict-free padding.
// LSTM h-broadcast: Tensor Data Mover (tensor_load_to_lds + s_wait_tensorcnt).
// ---------------------------------------------------------------------------

typedef __attribute__((ext_vector_type(2))) float        v2f;
typedef __attribute__((ext_vector_type(8))) float        v8f;
typedef __attribute__((ext_vector_type(4))) unsigned int v4u;
typedef __attribute__((ext_vector_type(8))) int          v8i;
typedef __attribute__((ext_vector_type(4))) int          v4i;

#define B_DIM   64
#define N_DIM   1024
#define H16     256
#define H64     1024
#define H128    2048
#define G4H     (4 * H128)   // 8192

#define KCHUNK  256
#define ASTRIDE 260          // KCHUNK + 4: stride % 64 == 4 -> conflict-free b64 frag reads

#if defined(__has_builtin)
#if __has_builtin(__builtin_amdgcn_tensor_load_to_lds)
#define HAVE_TDM 1
#endif
#endif

__device__ __forceinline__ float sigmoid_(float x) {
    return 1.0f / (1.0f + __expf(-x));
}

// ---------------------------------------------------------------------------
// Kernel 1: per-row GCN sampling (unchanged from round-1 version).
// ---------------------------------------------------------------------------
__global__ void gcn_kernel(const float* __restrict__ input,
                           const float* __restrict__ gc1_w,
                           const float* __restrict__ gc1_b,
                           const float* __restrict__ gc2_w,
                           const float* __restrict__ gc2_b,
                           float* __restrict__ xout)
{
    __shared__ float xs[N_DIM];
    __shared__ float s2[N_DIM];

    const int b   = blockIdx.x;
    const int tid = threadIdx.x;                 // 256 threads
    const float* row = input + (size_t)b * N_DIM;

    #pragma unroll
    for (int q = 0; q < 4; ++q) xs[tid + 256 * q] = row[tid + 256 * q];
    __syncthreads();

    const float g1w0 = gc1_w[0], g1w1 = gc1_w[1], g1w2 = gc1_w[2], g1w3 = gc1_w[3];
    const float g1b0 = gc1_b[0], g1b1 = gc1_b[1], g1b2 = gc1_b[2], g1b3 = gc1_b[3];
    const float g2w0 = gc2_w[0], g2w1 = gc2_w[1], g2w2 = gc2_w[2], g2w3 = gc2_w[3];
    const float g2b  = gc2_b[0];

    int nidx[4][4];

    #pragma unroll
    for (int q = 0; q < 4; ++q) {
        const int n  = tid + 256 * q;
        const float xn = xs[n];
        float d0 = INFINITY, d1 = INFINITY, d2 = INFINITY, d3 = INFINITY;
        int   i0 = 0, i1 = 0, i2 = 0, i3 = 0;
        for (int j = 0; j < N_DIM; ++j) {
            const float d = fabsf(xs[j] - xn);
            if (j == n) continue;                       // diagonal = +inf in reference
            if (d < d3) {
                if (d < d0)      { d3=d2;i3=i2; d2=d1;i2=i1; d1=d0;i1=i0; d0=d;i0=j; }
                else if (d < d1) { d3=d2;i3=i2; d2=d1;i2=i1; d1=d;i1=j; }
                else if (d < d2) { d3=d2;i3=i2; d2=d;i2=j; }
                else             { d3=d;i3=j; }
            }
        }
        nidx[q][0]=i0; nidx[q][1]=i1; nidx[q][2]=i2; nidx[q][3]=i3;

        const float S  = xs[i0] + xs[i1] + xs[i2] + xs[i3];
        const float h0 = fmaxf(fmaf(S, g1w0, g1b0), 0.0f);
        const float h1 = fmaxf(fmaf(S, g1w1, g1b1), 0.0f);
        const float h2 = fmaxf(fmaf(S, g1w2, g1b2), 0.0f);
        const float h3 = fmaxf(fmaf(S, g1w3, g1b3), 0.0f);
        s2[n] = h0*g2w0 + h1*g2w1 + h2*g2w2 + h3*g2w3;
    }
    __syncthreads();

    #pragma unroll
    for (int q = 0; q < 4; ++q) {
        const int n = tid + 256 * q;
        xout[(size_t)b * N_DIM + n] =
            s2[nidx[q][0]] + s2[nidx[q][1]] + s2[nidx[q][2]] + s2[nidx[q][3]] + g2b;
    }
}

// ---------------------------------------------------------------------------
// Kernel 2: fp32 WMMA GEMM:  out(64,N) = act( A(64,K) @ W(N,K)^T + bias [+bias2] )
// One wave per 16-wide N strip covering ALL of M (4 accumulators): each weight
// fragment feeds 4 v_wmma ops (4x weight-traffic reduction vs tile-per-wave).
// A is staged per-block in LDS in 64x256 K-chunks (stride 260 floats: a wave32
// ds_load_b64 fragment read hits banks {4l,4l+1,4l+2,4l+3}, all 64 distinct).
// EXEC is all-1s around every WMMA (no divergent control flow).
// ---------------------------------------------------------------------------
__global__ void gemm_wmma_kernel(const float* __restrict__ A,     // 64 x K row-major
                                 const float* __restrict__ W,     // N x K row-major
                                 const float* __restrict__ bias,  // N
                                 const float* __restrict__ bias2, // N or nullptr
                                 float* __restrict__ out,         // 64 x N
                                 int N, int Kd, int relu)
{
    __shared__ float as[64 * ASTRIDE];           // 66,560 B of the 320 KB WGP LDS

    const int tid  = threadIdx.x;                // 256 threads = 8 waves
    const int wave = tid >> 5;
    const int lane = tid & 31;
    const int tn   = blockIdx.x * 8 + wave;      // 16-wide output column strip
    const int mn   = lane & 15;                  // A row within tile / B col
    const int kh   = lane >> 4;                  // K-pair select

    const float* Wbase = W + (size_t)(tn * 16 + mn) * Kd + 2 * kh;

    v8f acc[4] = {};

    for (int kbase = 0; kbase < Kd; kbase += KCHUNK) {
        __syncthreads();                         // retire previous chunk's readers
        // Cooperative stage of A chunk: 64 rows x 256 cols as 4096 float4s.
        #pragma unroll
        for (int i = 0; i < 16; ++i) {
            const int f   = tid + 256 * i;       // float4 index
            const int row = f >> 6;
            const int c4  = f & 63;
            const float4 v = *(const float4*)(A + (size_t)row * Kd + kbase + c4 * 4);
            *(float4*)(&as[row * ASTRIDE + c4 * 4]) = v;   // 16B-aligned (260 % 4 == 0)
        }
        __syncthreads();

        const float* wp = Wbase + kbase;
        for (int kk = 0; kk < KCHUNK; kk += 4) {
            const v2f b  = *(const v2f*)(wp + kk);         // weight frag: load once...
            const int la = kk + 2 * kh;
            #pragma unroll
            for (int tm = 0; tm < 4; ++tm) {               // ...use 4 times
                const v2f a = *(const v2f*)(&as[(tm * 16 + mn) * ASTRIDE + la]);
                acc[tm] = __builtin_amdgcn_wmma_f32_16x16x4_f32(
                              false, a, false, b, (short)0, acc[tm], false, false);
            }
        }
    }

    float bn = bias[tn * 16 + mn];
    if (bias2) bn += bias2[tn * 16 + mn];

    #pragma unroll
    for (int tm = 0; tm < 4; ++tm) {
        #pragma unroll
        for (int r = 0; r < 8; ++r) {
            const int rowm = tm * 16 + r + 8 * kh;         // C/D frag layout
            float v = acc[tm][r] + bn;
            if (relu) v = fmaxf(v, 0.0f);
            out[(size_t)rowm * N + tn * 16 + mn] = v;
        }
    }
}

// ---------------------------------------------------------------------------
// Kernel 3: LSTM recurrent matvec  g_h(8192) = h(2048) @ w_hh(8192,2048)^T.
// h (8 KB) is broadcast into LDS by the Tensor Data Mover: wave 0 builds a D#
// (count=1, type=2 "image", data_size=4B, 1-row tile of 2048 elements) and
// issues tensor_load_to_lds, waits on TENSORcnt, then the workgroup barrier
// publishes the LDS tile. w_hh is L2-resident after step 0 (64 MB < 192 MB L2).
// ---------------------------------------------------------------------------
__global__ void lstm_matvec_kernel(const float* __restrict__ w_hh,  // 8192 x 2048
                                   const float* __restrict__ h,     // 2048
                                   float* __restrict__ g_h)         // 8192
{
    __shared__ float hs[H128];
    const int tid  = threadIdx.x;                 // 256 threads
    const int wave = tid >> 5;
    const int lane = tid & 31;

#if defined(HAVE_TDM)
    if (wave == 0) {
        // LDS aperture maps generic addr[31:0] -> LDS byte offset (ISA 10.2).
        const unsigned lds_addr = (unsigned)(size_t)(&hs[0]);
        const unsigned long long ga = (unsigned long long)(size_t)h;

        v4u g0;
        g0[0] = 1u;                                            // count=1, user mode
        g0[1] = lds_addr;                                      // lds_addr
        g0[2] = (unsigned)(ga & 0xFFFFFFFFu);                  // global_addr[31:0]
        g0[3] = (unsigned)((ga >> 32) & 0x1FFFFFFu)            // global_addr[56:32]
              | (2u << 30);                                    // type=2 ("image")

        v8i g1 = {};
        g1[0] = (int)(2u << 16);          // data_size=2 (4 bytes); workgroup_mask=0
        g1[1] = (int)(2048u << 16);       // tensor_dim0[15:0]  -> bits 63:48
        g1[2] = (int)(1u << 16);          // tensor_dim0[31:16]=0; tensor_dim1=1
        g1[3] = (int)(2048u << 16);       // tile_dim0 = 2048   -> bits 127:112
        g1[4] = 1;                        // tile_dim1 = 1
        g1[5] = 2048;                     // tensor_dim0_stride[31:0]
        g1[6] = 0;
        g1[7] = 0;

        v4i gz = {};
#if __clang_major__ >= 23
        __builtin_amdgcn_tensor_load_to_lds(g0, g1, gz, gz, g1, 0);
#else
        __builtin_amdgcn_tensor_load_to_lds(g0, g1, gz, gz, 0);
#endif
#if __has_builtin(__builtin_amdgcn_s_wait_tensorcnt)
        __builtin_amdgcn_s_wait_tensorcnt(0);
#else
        asm volatile("s_wait_tensorcnt 0x0" ::: "memory");
#endif
    }
#else
    #pragma unroll
    for (int q = 0; q < 8; ++q) hs[tid + 256 * q] = h[tid + 256 * q];
#endif
    __syncthreads();

    const int j = blockIdx.x * 8 + wave;          // output index (0..8191)
    const float* wr = w_hh + (size_t)j * H128;

    float acc = 0.0f;
    #pragma unroll 4
    for (int i = 0; i < 16; ++i) {
        const int idx = lane * 4 + 128 * i;
        __builtin_prefetch(wr + idx + 512, 0, 0);           // global_prefetch_b8
        const float4 w4 = *(const float4*)(wr + idx);
        const float4 h4 = *(const float4*)(&hs[idx]);
        acc += w4.x * h4.x + w4.y * h4.y + w4.z * h4.z + w4.w * h4.w;
    }
    #pragma unroll
    for (int off = 16; off > 0; off >>= 1)
        acc += __shfl_xor(acc, off, 32);
    if (lane == 0) g_h[j] = acc;
}

// ---------------------------------------------------------------------------
// Kernel 4: LSTM gate update for one timestep (2048 threads).
// ---------------------------------------------------------------------------
__global__ void lstm_gate_kernel(const float* __restrict__ Gpre_t, // 8192
                                 const float* __restrict__ g_h,    // 8192
                                 float* __restrict__ c,            // 2048 (in/out)
                                 float* __restrict__ h,            // 2048 (in/out)
                                 float* __restrict__ out_t)        // 2048
{
    const int u = blockIdx.x * blockDim.x + threadIdx.x;
    const float ig = Gpre_t[u]            + g_h[u];
    const float fg = Gpre_t[H128 + u]     + g_h[H128 + u];
    const float gg = Gpre_t[2 * H128 + u] + g_h[2 * H128 + u];
    const float og = Gpre_t[3 * H128 + u] + g_h[3 * H128 + u];
    const float cn = sigmoid_(fg) * c[u] + sigmoid_(ig) * tanhf(gg);
    const float hn = sigmoid_(og) * tanhf(cn);
    c[u] = cn;
    h[u] = hn;
    out_t[u] = hn;
}

__global__ void zero_kernel(float* __restrict__ p, int n)
{
    const int i = blockIdx.x * blockDim.x + threadIdx.x;
    if (i < n) p[i] = 0.0f;
}

// ---------------------------------------------------------------------------
// Host-side launch sequence (graph-capture safe: only kernel launches).
// ---------------------------------------------------------------------------
extern "C" void kernel_launch(void* const* d_in, const int* in_sizes, int n_in,
                              void* d_out, int out_size, void* d_ws, size_t ws_size,
                              hipStream_t stream) {
    (void)in_sizes; (void)n_in; (void)out_size; (void)ws_size;

    const float* input = (const float*)d_in[0];
    // d_in[1]=Hidden_State, d_in[2]=Cell_State: reference ignores them (h0 = zeros)
    const float* gc1_w = (const float*)d_in[3];
    const float* gc1_b = (const float*)d_in[4];
    const float* gc2_w = (const float*)d_in[5];
    const float* gc2_b = (const float*)d_in[6];
    const float* fl_w  = (const float*)d_in[7];
    const float* fl_b  = (const float*)d_in[8];
    const float* il_w  = (const float*)d_in[9];
    const float* il_b  = (const float*)d_in[10];
    const float* ol_w  = (const float*)d_in[11];
    const float* ol_b  = (const float*)d_in[12];
    const float* w_ih  = (const float*)d_in[13];
    const float* w_hh  = (const float*)d_in[14];
    const float* b_ih  = (const float*)d_in[15];
    const float* b_hh  = (const float*)d_in[16];

    float* ws    = (float*)d_ws;
    float* x_gcn = ws;                         // 64*1024
    float* f1    = x_gcn + B_DIM * N_DIM;      // 64*256
    float* f2    = f1 + B_DIM * H16;           // 64*1024
    float* z     = f2 + B_DIM * H64;           // 64*2048
    float* Gpre  = z + B_DIM * H128;           // 64*8192
    float* g_h   = Gpre + B_DIM * G4H;         // 8192
    float* hbuf  = g_h + G4H;                  // 2048
    float* cbuf  = hbuf + H128;                // 2048

    // 1) GCN front-end: (64,1024) -> (64,1024)
    gcn_kernel<<<B_DIM, 256, 0, stream>>>(input, gc1_w, gc1_b, gc2_w, gc2_b, x_gcn);

    // 2) MLP chain, fp32 WMMA, relu fused. Grid = (N/16)/8 blocks (8 waves/block,
    //    each wave owns a full-M 16-col strip).
    gemm_wmma_kernel<<<(H16  / 16) / 8, 256, 0, stream>>>(x_gcn, fl_w, fl_b, nullptr, f1, H16,  N_DIM, 1);
    gemm_wmma_kernel<<<(H64  / 16) / 8, 256, 0, stream>>>(f1,    il_w, il_b, nullptr, f2, H64,  H16,  1);
    gemm_wmma_kernel<<<(H128 / 16) / 8, 256, 0, stream>>>(f2,    ol_w, ol_b, nullptr, z,  H128, H64,  1);

    // 3) Batched non-recurrent LSTM projection for all 64 timesteps:
    //    Gpre(64,8192) = z @ w_ih^T + (b_ih + b_hh)
    gemm_wmma_kernel<<<(G4H / 16) / 8, 256, 0, stream>>>(z, w_ih, b_ih, b_hh, Gpre, G4H, H128, 0);

    // 4) LSTM recurrence: h0 = c0 = 0, then 64 sequential (matvec, gate) pairs.
    zero_kernel<<<(2 * H128 + 255) / 256, 256, 0, stream>>>(hbuf, 2 * H128);
    float* out = (float*)d_out;
    for (int t = 0; t < B_DIM; ++t) {
        lstm_matvec_kernel<<<G4H / 8, 256, 0, stream>>>(w_hh, hbuf, g_h);
        lstm_gate_kernel<<<H128 / 256, 256, 0, stream>>>(Gpre + (size_t)t * G4H, g_h,
                                                         cbuf, hbuf, out + (size_t)t * H128);
    }
}